// ComplexBatchNorm2d_43499428774583
// MI455X (gfx1250) — compile-verified
//
#include <hip/hip_runtime.h>

typedef __attribute__((ext_vector_type(2))) float v2f;
typedef __attribute__((ext_vector_type(4))) float v4f;
typedef __attribute__((ext_vector_type(8))) float v8f;

#define B_      32
#define C_      64
#define HW_     16384           // 128*128
#define NPLANE4 4096            // HW_/4 float4 per plane
#define NBLK    2048            // B_*C_
#define EPS_    1e-5f

// Full 32-lane sum through the matrix unit:
// A[m][0..1]=v_m, A[m][2..3]=v_{m+16}; B=ones(4x16) -> D[m][n]=2(v_m+v_{m+16}).
// Per-lane sum of the 8 D VGPRs gives half-sums; xor-16 shuffle combines halves.
__device__ __forceinline__ float wave_sum32(float v) {
  v2f a;  a[0] = v;    a[1] = v;
  v2f bo; bo[0] = 1.f; bo[1] = 1.f;
  v8f c = {};
  v8f d = __builtin_amdgcn_wmma_f32_16x16x4_f32(false, a, false, bo,
                                                (short)0, c, false, false);
  float s = ((d[0] + d[1]) + (d[2] + d[3])) + ((d[4] + d[5]) + (d[6] + d[7]));
  s += __shfl_xor(s, 16, 32);
  return 0.5f * s;
}

// Pass 1: per-(b,c) plane-pair partial sums -> ws
__global__ __launch_bounds__(256) void cbn_reduce(const float* __restrict__ x,
                                                  float* __restrict__ part) {
  const int blk = blockIdx.x;
  const int b = blk >> 6;
  const int c = blk & 63;
  const v4f* __restrict__ xr =
      (const v4f*)(x + (size_t)b * (size_t)(2 * C_ * HW_) + (size_t)c * HW_);
  const v4f* __restrict__ xi = xr + (C_ * HW_ / 4);
  const int tid = threadIdx.x;

  float sr = 0.f, si = 0.f, srr = 0.f, sii = 0.f, sri = 0.f;
#pragma unroll 4
  for (int j = 0; j < NPLANE4; j += 256) {
    if (j + 2048 < NPLANE4) {
      __builtin_prefetch(&xr[j + tid + 2048], 0, 1);
      __builtin_prefetch(&xi[j + tid + 2048], 0, 1);
    }
    v4f r = xr[j + tid];
    v4f i = xi[j + tid];
    sr  += (r[0] + r[1]) + (r[2] + r[3]);
    si  += (i[0] + i[1]) + (i[2] + i[3]);
    srr += (r[0] * r[0] + r[1] * r[1]) + (r[2] * r[2] + r[3] * r[3]);
    sii += (i[0] * i[0] + i[1] * i[1]) + (i[2] * i[2] + i[3] * i[3]);
    sri += (r[0] * i[0] + r[1] * i[1]) + (r[2] * i[2] + r[3] * i[3]);
  }

  sr  = wave_sum32(sr);
  si  = wave_sum32(si);
  srr = wave_sum32(srr);
  sii = wave_sum32(sii);
  sri = wave_sum32(sri);

  __shared__ float lds[8][5];
  const int wv = tid >> 5;
  if ((tid & 31) == 0) {
    lds[wv][0] = sr; lds[wv][1] = si; lds[wv][2] = srr;
    lds[wv][3] = sii; lds[wv][4] = sri;
  }
  __syncthreads();
  if (tid < 5) {
    float s = 0.f;
#pragma unroll
    for (int w = 0; w < 8; ++w) s += lds[w][tid];
    part[(size_t)blk * 8 + tid] = s;
  }
}

// Pass 2: per-channel stats -> fused affine coefficients (exactly mirrors reference math)
__global__ __launch_bounds__(64) void cbn_stats(const float* __restrict__ part,
                                                const float* __restrict__ gamma,
                                                const float* __restrict__ beta,
                                                float* __restrict__ coef) {
  const int c = threadIdx.x;  // one channel per lane, 64 threads
  float sr = 0.f, si = 0.f, srr = 0.f, sii = 0.f, sri = 0.f;
  for (int b = 0; b < B_; ++b) {
    const float* p = part + (size_t)(b * C_ + c) * 8;
    sr += p[0]; si += p[1]; srr += p[2]; sii += p[3]; sri += p[4];
  }
  const float n    = (float)(B_ * HW_);        // 524288
  const float dinv = 1.0f / (float)(B_ * HW_ - 1);
  const float mr = sr / n, mi = si / n;
  const float covRR = (srr - n * mr * mr) * dinv;
  const float covII = (sii - n * mi * mi) * dinv;
  const float covRI = (sri - n * mr * mi) * dinv;

  const float m00 = covRR + EPS_, m11 = covII + EPS_;
  const float m01 = covRI, m10 = covRI;
  const float tau   = m00 + m11;
  const float delta = m00 * m11 - m01 * m10;
  const float s = sqrtf(delta);
  const float t = sqrtf(tau + 2.0f * delta);   // reference uses tau + 2*delta
  const float rt = 1.0f / t;
  const float q00 = (m00 + s) * rt, q01 = (m01 + s) * rt;
  const float q10 = (m10 + s) * rt, q11 = (m11 + s) * rt;
  const float det  = q00 * q11 - q01 * q10;
  const float idet = 1.0f / det;
  const float iv00 =  q11 * idet, iv01 = -q01 * idet;
  const float iv10 = -q10 * idet, iv11 =  q00 * idet;

  const float g00 = gamma[c * 4 + 0], g01 = gamma[c * 4 + 1];
  const float g10 = gamma[c * 4 + 2], g11 = gamma[c * 4 + 3];
  const float a00 = g00 * iv00 + g01 * iv10;
  const float a01 = g00 * iv01 + g01 * iv11;
  const float a10 = g10 * iv00 + g11 * iv10;
  const float a11 = g10 * iv01 + g11 * iv11;
  const float b0  = beta[c * 2 + 0], b1 = beta[c * 2 + 1];

  float* o = coef + c * 8;
  o[0] = a00; o[1] = a01; o[2] = a10; o[3] = a11;
  o[4] = b0 - a00 * mr - a01 * mi;   // fold mean subtraction into bias
  o[5] = b1 - a10 * mr - a11 * mi;
}

// Pass 3: fused per-channel 2x2 affine; reverse plane order for L2 reuse,
// nontemporal stores for the write-once output.
__global__ __launch_bounds__(256) void cbn_apply(const float* __restrict__ x,
                                                 const float* __restrict__ coef,
                                                 float* __restrict__ y) {
  const int blk = (NBLK - 1) - (int)blockIdx.x;
  const int b = blk >> 6;
  const int c = blk & 63;
  const size_t off = (size_t)b * (size_t)(2 * C_ * HW_) + (size_t)c * HW_;
  const v4f* __restrict__ xr = (const v4f*)(x + off);
  const v4f* __restrict__ xi = xr + (C_ * HW_ / 4);
  v4f* __restrict__ yr = (v4f*)(y + off);
  v4f* __restrict__ yi = yr + (C_ * HW_ / 4);

  const float* A = coef + c * 8;
  const float a00 = A[0], a01 = A[1], a10 = A[2], a11 = A[3];
  const float b0 = A[4], b1 = A[5];
  const int tid = threadIdx.x;

#pragma unroll 4
  for (int j = 0; j < NPLANE4; j += 256) {
    v4f r = xr[j + tid];
    v4f i = xi[j + tid];
    v4f orv, oiv;
#pragma unroll
    for (int k = 0; k < 4; ++k) {
      orv[k] = fmaf(a00, r[k], fmaf(a01, i[k], b0));
      oiv[k] = fmaf(a10, r[k], fmaf(a11, i[k], b1));
    }
    __builtin_nontemporal_store(orv, &yr[j + tid]);
    __builtin_nontemporal_store(oiv, &yi[j + tid]);
  }
}

extern "C" void kernel_launch(void* const* d_in, const int* in_sizes, int n_in,
                              void* d_out, int out_size, void* d_ws, size_t ws_size,
                              hipStream_t stream) {
  const float* x     = (const float*)d_in[0];
  const float* gamma = (const float*)d_in[1];
  const float* beta  = (const float*)d_in[2];
  float* y    = (float*)d_out;
  float* part = (float*)d_ws;            // NBLK*8 floats = 64 KB
  float* coef = part + (size_t)NBLK * 8; // 64*8 floats

  cbn_reduce<<<NBLK, 256, 0, stream>>>(x, part);
  cbn_stats<<<1, 64, 0, stream>>>(part, gamma, beta, coef);
  cbn_apply<<<NBLK, 256, 0, stream>>>(x, coef, y);
}